// RotaryPositionEmbedding_51161650430263
// MI455X (gfx1250) — compile-verified
//
#include <hip/hip_runtime.h>

// RoPE forward, MI455X (gfx1250), wave32.
// x: (8,16,4096,64) fp32; token_positions: (8,16,4096) int32 (JAX x64 off).
// Memory-bound: ~258 MiB @ 23.3 TB/s ~= 11.2 us floor.
//
// Mapping: each thread owns one float4 slot q (= head-dim pairs 2q,2q+1) and
// walks ROWS_PER_THREAD token rows strided by num_groups. inv_freq (v_exp_f32)
// is lane-constant and computed once; each step is perfectly coalesced
// (lanes 0..15 -> row r, lanes 16..31 -> row r+1 => 512 B contiguous / wave).

typedef __attribute__((ext_vector_type(4))) float v4f;

// ln(10000)/32 : inv_freq[j] = 10000^(-j/32) = exp(-j * ln(10000)/32), j=0..31
#define LN_THETA_OVER_HALF 0.28782313662425572f
#define ROWS_PER_THREAD 4u

__global__ __launch_bounds__(256, 1)
void rope_fwd_f32(const float* __restrict__ x,
                  const int*   __restrict__ tokpos,
                  float*       __restrict__ out,
                  unsigned     num_groups)   // total_rows / ROWS_PER_THREAD
{
  const unsigned t = blockIdx.x * 256u + threadIdx.x;
  const unsigned q = t & 15u;   // float4 slot within the 64-wide head dim
  const unsigned G = t >> 4;    // row group; rows handled: G + k*num_groups

  // Lane-constant inverse frequencies for pairs j0=2q, j1=2q+1 — two
  // v_exp_f32, amortized over ROWS_PER_THREAD rows.
  const float j0 = (float)(2u * q);
  const float f0 = __expf(-(j0)        * LN_THETA_OVER_HALF);
  const float f1 = __expf(-(j0 + 1.0f) * LN_THETA_OVER_HALF);

  // Unguarded speculative stream prefetch ~1 MiB ahead (global_prefetch_b8,
  // SYS scope -> GL2 only; OOB translations are silently dropped, no branch).
  __builtin_prefetch((const void*)((const v4f*)x + ((size_t)G * 16u + q + (16u * 4096u))), 0, 0);

#pragma unroll
  for (unsigned k = 0; k < ROWS_PER_THREAD; ++k) {
    const unsigned row = G + k * num_groups;
    const size_t   e   = (size_t)row * 16u + q;   // float4 index

    // Position uniform across each 16-lane half-wave -> one broadcast fetch.
    const float p  = (float)tokpos[row];
    const float a0 = p * f0;
    const float a1 = p * f1;

    float s0, c0, s1, c1;
    __sincosf(a0, &s0, &c0);   // native v_sin_f32 / v_cos_f32
    __sincosf(a1, &s1, &c1);

    // Touch-once streams: NT B128 keeps 256 MiB of single-use data from
    // churning the 192 MB L2.
    v4f v = __builtin_nontemporal_load((const v4f*)x + e);

    v4f o;
    o.x = fmaf(c0, v.x, -(s0 * v.y));   // cos*x1 - sin*x2
    o.y = fmaf(s0, v.x,  (c0 * v.y));   // sin*x1 + cos*x2
    o.z = fmaf(c1, v.z, -(s1 * v.w));
    o.w = fmaf(s1, v.z,  (c1 * v.w));

    __builtin_nontemporal_store(o, (v4f*)out + e);
  }
}

extern "C" void kernel_launch(void* const* d_in, const int* in_sizes, int n_in,
                              void* d_out, int out_size, void* d_ws, size_t ws_size,
                              hipStream_t stream) {
  const float* x      = (const float*)d_in[0];
  const int*   tokpos = (const int*)  d_in[1];
  float*       out    = (float*)d_out;

  const unsigned rows       = (unsigned)in_sizes[1];        // 8*16*4096 = 524288
  const unsigned num_groups = rows / ROWS_PER_THREAD;       // 131072
  const unsigned threads    = 256;                          // 8 wave32 waves
  const unsigned blocks     = (num_groups * 16u) / threads; // 8192

  rope_fwd_f32<<<blocks, threads, 0, stream>>>(x, tokpos, out, num_groups);
}